// MaskedMaxPool_8521215115543
// MI455X (gfx1250) — compile-verified
//
#include <hip/hip_runtime.h>

#define NBATCH 8
#define NPTS   16384
#define NFEAT  128
#define NPOINT 512
#define KNN    32
#define R2     0.04f

typedef __attribute__((ext_vector_type(2))) float v2f;
typedef __attribute__((ext_vector_type(8))) float v8f;

// ---------------------------------------------------------------------------
// Kernel 1: farthest point sampling. One 1024-thread workgroup per batch.
// Each thread keeps 16 points (coords + running min-distance) in registers
// for all 512 iterations. Writes new_xyz (output 0) directly.
// ---------------------------------------------------------------------------
__global__ __launch_bounds__(1024) void fps_kernel(const float* __restrict__ xyz,
                                                   float* __restrict__ new_xyz) {
  const int b = blockIdx.x;
  const int t = threadIdx.x;
  const int lane = t & 31;
  const int wid  = t >> 5;
  const float* bx = xyz + (size_t)b * NPTS * 3;

  float px[16], py[16], pz[16], dist[16];
#pragma unroll
  for (int j = 0; j < 16; ++j) {
    int n = j * 1024 + t;
    px[j] = bx[n * 3 + 0];
    py[j] = bx[n * 3 + 1];
    pz[j] = bx[n * 3 + 2];
    dist[j] = 1e10f;
  }

  __shared__ float s_cx, s_cy, s_cz;
  __shared__ int   s_far;
  __shared__ float s_wv[32];
  __shared__ int   s_wi[32];

  int far = 0;
  for (int it = 0; it < NPOINT; ++it) {
    // owner thread publishes centroid coords (far is uniform)
    int oj = far >> 10;
    int ot = far & 1023;
    if (t == ot) { s_cx = px[oj]; s_cy = py[oj]; s_cz = pz[oj]; }
    __syncthreads();
    float cx = s_cx, cy = s_cy, cz = s_cz;
    if (t == 0) {
      float* o = new_xyz + ((size_t)b * NPOINT + it) * 3;
      o[0] = cx; o[1] = cy; o[2] = cz;
    }

    float bv = -1.0f; int bi = 0;
#pragma unroll
    for (int j = 0; j < 16; ++j) {
      float dx = px[j] - cx, dy = py[j] - cy, dz = pz[j] - cz;
      float d = dx * dx + dy * dy + dz * dz;
      float nd = fminf(dist[j], d);
      dist[j] = nd;
      int n = j * 1024 + t;
      if (nd > bv) { bv = nd; bi = n; }   // ascending n => '>' keeps first max
    }
    // wave32 argmax reduce (tie -> smaller index, matching argmax-first)
#pragma unroll
    for (int off = 16; off > 0; off >>= 1) {
      float ov = __shfl_xor(bv, off, 32);
      int   oi = __shfl_xor(bi, off, 32);
      if (ov > bv || (ov == bv && oi < bi)) { bv = ov; bi = oi; }
    }
    if (lane == 0) { s_wv[wid] = bv; s_wi[wid] = bi; }
    __syncthreads();
    if (wid == 0) {
      bv = s_wv[lane]; bi = s_wi[lane];
#pragma unroll
      for (int off = 16; off > 0; off >>= 1) {
        float ov = __shfl_xor(bv, off, 32);
        int   oi = __shfl_xor(bi, off, 32);
        if (ov > bv || (ov == bv && oi < bi)) { bv = ov; bi = oi; }
      }
      if (lane == 0) s_far = bi;
    }
    __syncthreads();
    far = s_far;
  }
}

// ---------------------------------------------------------------------------
// Kernel 2: ball query. One 512-thread (16-wave) workgroup per 16 centroids.
// Distance tiles via V_WMMA_F32_16X16X4_F32:
//   D = (-2*C[16x3,pad])*(P[3x16,pad]) + (|c|^2 + |p|^2)
// Each wave then scans its centroid's distances in index order with
// ballot+prefix append (first 32 hits), with collective early exit.
// ---------------------------------------------------------------------------
__global__ __launch_bounds__(512) void ballq_kernel(const float* __restrict__ xyz,
                                                    const float* __restrict__ new_xyz,
                                                    int* __restrict__ idx) {
  const int b    = blockIdx.y;
  const int c0   = blockIdx.x * 16;
  const int t    = threadIdx.x;
  const int lane = t & 31;
  const int w    = t >> 5;               // wave id 0..15
  const int half = (lane >= 16) ? 1 : 0;
  const int m    = lane & 15;

  __shared__ float s_cx[16], s_cy[16], s_cz[16], s_cn2[16];
  __shared__ float s_dist[16 * 260];     // stride 260 to dodge bank conflicts
  __shared__ int   s_notDone;

  if (t == 0) s_notDone = 16;
  if (t < 16) {
    const float* c = new_xyz + ((size_t)b * NPOINT + c0 + t) * 3;
    float x = c[0], y = c[1], z = c[2];
    s_cx[t] = x; s_cy[t] = y; s_cz[t] = z;
    s_cn2[t] = x * x + y * y + z * z;
  }
  __syncthreads();

  // A matrix (16x4, per-lane 2 regs): lanes<16 hold K=0,1 ; lanes>=16 hold K=2,3
  v2f av;
  av.x = half ? (-2.0f * s_cz[m]) : (-2.0f * s_cx[m]);
  av.y = half ? 0.0f               : (-2.0f * s_cy[m]);
  // C layout: VGPR i -> row M = i + 8*half
  float cn[8];
#pragma unroll
  for (int i = 0; i < 8; ++i) cn[i] = s_cn2[i + 8 * half];

  const float* bx = xyz + (size_t)b * NPTS * 3;
  int* myIdx = idx + ((size_t)b * NPOINT + c0 + w) * KNN;

  int cnt = 0, first = 0;
  bool done = false;

  for (int pass = 0; pass < NPTS / 256; ++pass) {
    // ---- compute phase (all threads active: EXEC all ones for WMMA) ----
    int n = pass * 256 + w * 16 + m;
    float qx = bx[n * 3 + 0], qy = bx[n * 3 + 1], qz = bx[n * 3 + 2];
    float pn2 = qx * qx + qy * qy + qz * qz;
    v2f bvv;
    bvv.x = half ? qz : qx;
    bvv.y = half ? 0.0f : qy;
    v8f cv;
#pragma unroll
    for (int i = 0; i < 8; ++i) cv[i] = cn[i] + pn2;
    v8f dv = __builtin_amdgcn_wmma_f32_16x16x4_f32(false, av, false, bvv,
                                                   (short)0, cv, false, false);
#pragma unroll
    for (int i = 0; i < 8; ++i)
      s_dist[(i + 8 * half) * 260 + w * 16 + m] = dv[i];
    __syncthreads();

    // ---- scan phase: wave w scans centroid (c0 + w), index order ----
    if (!done) {
#pragma unroll
      for (int chunk = 0; chunk < 8; ++chunk) {
        float dd = s_dist[w * 260 + chunk * 32 + lane];
        int   ng = pass * 256 + chunk * 32 + lane;
        bool pred = (dd <= R2);
        unsigned mask = __builtin_amdgcn_ballot_w32(pred);
        if (mask) {
          if (cnt == 0) first = pass * 256 + chunk * 32 + (__ffs((int)mask) - 1);
          int prefix = __popc(mask & ((1u << lane) - 1u));
          int pos = cnt + prefix;
          if (pred && pos < KNN) myIdx[pos] = ng;
          cnt += __popc(mask);
          if (cnt >= KNN) {
            done = true;
            if (lane == 0) atomicSub(&s_notDone, 1);
            break;
          }
        }
      }
    }
    __syncthreads();
    if (s_notDone == 0) break;           // uniform
  }

  // pad remaining slots with first hit (reference: idx==N -> first)
  if (lane < KNN && lane >= cnt) myIdx[lane] = first;
}

// ---------------------------------------------------------------------------
// Kernel 3: transpose features (B,128,N) -> (B,N,128) for coalesced gathers.
// ---------------------------------------------------------------------------
__global__ __launch_bounds__(256) void transpose_kernel(const float* __restrict__ feat,
                                                        float* __restrict__ featT) {
  __shared__ float tile[32][33];
  const int b  = blockIdx.z;
  const int f0 = blockIdx.y * 32;
  const int n0 = blockIdx.x * 32;
  const int x  = threadIdx.x;            // 0..31
  const int y0 = threadIdx.y;            // 0..7
  const float* fb = feat  + (size_t)b * NFEAT * NPTS;
  float*       tb = featT + (size_t)b * NPTS * NFEAT;
#pragma unroll
  for (int y = y0; y < 32; y += 8)
    tile[y][x] = fb[(size_t)(f0 + y) * NPTS + n0 + x];
  __syncthreads();
#pragma unroll
  for (int y = y0; y < 32; y += 8)
    tb[(size_t)(n0 + y) * NFEAT + f0 + x] = tile[x][y];
}

// ---------------------------------------------------------------------------
// Kernel 4: grouped max-pool, transposed-feature (coalesced) path.
// ---------------------------------------------------------------------------
__global__ __launch_bounds__(128) void maxpool_t_kernel(const float* __restrict__ featT,
                                                        const int* __restrict__ idx,
                                                        float* __restrict__ out) {
  const int b = blockIdx.x / NPOINT;
  const int s = blockIdx.x % NPOINT;
  const int f = threadIdx.x;
  __shared__ int nid[KNN];
  if (f < KNN) nid[f] = idx[((size_t)b * NPOINT + s) * KNN + f];
  __syncthreads();
  const float* tb = featT + (size_t)b * NPTS * NFEAT;
  float mx = -3.4e38f;
#pragma unroll 4
  for (int k = 0; k < KNN; ++k)
    mx = fmaxf(mx, tb[(size_t)nid[k] * NFEAT + f]);
  out[((size_t)b * NFEAT + f) * NPOINT + s] = mx;
}

// Fallback: gather directly from (B,128,N) layout (L2-resident, uncoalesced).
__global__ __launch_bounds__(128) void maxpool_d_kernel(const float* __restrict__ feat,
                                                        const int* __restrict__ idx,
                                                        float* __restrict__ out) {
  const int b = blockIdx.x / NPOINT;
  const int s = blockIdx.x % NPOINT;
  const int f = threadIdx.x;
  __shared__ int nid[KNN];
  if (f < KNN) nid[f] = idx[((size_t)b * NPOINT + s) * KNN + f];
  __syncthreads();
  const float* fb = feat + ((size_t)b * NFEAT + f) * NPTS;
  float mx = -3.4e38f;
#pragma unroll 4
  for (int k = 0; k < KNN; ++k)
    mx = fmaxf(mx, fb[nid[k]]);
  out[((size_t)b * NFEAT + f) * NPOINT + s] = mx;
}

// ---------------------------------------------------------------------------
extern "C" void kernel_launch(void* const* d_in, const int* in_sizes, int n_in,
                              void* d_out, int out_size, void* d_ws, size_t ws_size,
                              hipStream_t stream) {
  (void)in_sizes; (void)n_in; (void)out_size;
  const float* xyz      = (const float*)d_in[0];   // (8,16384,3)
  const float* features = (const float*)d_in[1];   // (8,128,16384)

  float* out     = (float*)d_out;
  float* new_xyz = out;                                  // (8,512,3)
  float* subfeat = out + (size_t)NBATCH * NPOINT * 3;    // (8,128,512)

  int*   idx   = (int*)d_ws;                                      // 512 KB
  float* featT = (float*)((char*)d_ws + (size_t)(1 << 20));       // 64 MB
  const size_t needT = (size_t)(1 << 20) + (size_t)NBATCH * NPTS * NFEAT * 4;

  fps_kernel<<<NBATCH, 1024, 0, stream>>>(xyz, new_xyz);

  ballq_kernel<<<dim3(NPOINT / 16, NBATCH), 512, 0, stream>>>(xyz, new_xyz, idx);

  if (ws_size >= needT) {
    transpose_kernel<<<dim3(NPTS / 32, NFEAT / 32, NBATCH), dim3(32, 8), 0, stream>>>(
        features, featT);
    maxpool_t_kernel<<<NBATCH * NPOINT, 128, 0, stream>>>(featT, idx, subfeat);
  } else {
    maxpool_d_kernel<<<NBATCH * NPOINT, 128, 0, stream>>>(features, idx, subfeat);
  }
}